// GraphSAGEModel_29308856828498
// MI455X (gfx1250) — compile-verified
//
#include <hip/hip_runtime.h>

#define NN      50000
#define DIM     128
#define HOUT_HD 64
#define NEDGE   800000
#define BN_EPS  1e-5f

typedef __attribute__((ext_vector_type(2))) float v2f;
typedef __attribute__((ext_vector_type(4))) float v4f;
typedef __attribute__((ext_vector_type(8))) float v8f;

// ---------------------------------------------------------------- utilities
__global__ void zero_f32(float* __restrict__ p, int n) {
  int i = blockIdx.x * blockDim.x + threadIdx.x;
  if (i < n) p[i] = 0.0f;
}

__global__ void count_deg(const int* __restrict__ dst, int* __restrict__ cnt, int e) {
  int i = blockIdx.x * blockDim.x + threadIdx.x;
  if (i < e) atomicAdd(&cnt[dst[i]], 1);
}

__global__ void inv_deg(const int* __restrict__ cnt, float* __restrict__ inv, int n) {
  int i = blockIdx.x * blockDim.x + threadIdx.x;
  if (i < n) {
    int c = cnt[i];
    inv[i] = 1.0f / (float)(c > 0 ? c : 1);
  }
}

// --------------------------------------------------------- edge scatter-add
// one wave per edge; lane covers 4 consecutive channels (32*4 = 128)
__global__ void scatter_add(const float* __restrict__ x,
                            const int* __restrict__ src,
                            const int* __restrict__ dst,
                            float* __restrict__ agg, int e) {
  int wid  = (blockIdx.x * blockDim.x + threadIdx.x) >> 5;
  int lane = threadIdx.x & 31;
  if (wid >= e) return;
  int s = src[wid];
  int d = dst[wid];
  v4f v = *(const v4f*)(x + (size_t)s * DIM + lane * 4);
  float* ad = agg + (size_t)d * DIM + lane * 4;
  unsafeAtomicAdd(ad + 0, v.x);
  unsafeAtomicAdd(ad + 1, v.y);
  unsafeAtomicAdd(ad + 2, v.z);
  unsafeAtomicAdd(ad + 3, v.w);
}

// ------------------------------------------------------------- WMMA GEMM
// Y[m,n] = sum_k A1s[m,k]*W1[n,k]  (+ sum_k A2[m,k]*W2[n,k])  + bias[n]
// A1s = A1 * rowscale (mean aggregation folded in at LDS-stage time).
// Block: (HOUT/16) waves; each wave owns one 16x16 output tile.
// K fixed at DIM=128 -> 32 (or 64) chained v_wmma_f32_16x16x4_f32 per wave.
template <int HOUT, bool TWO>
__global__ __launch_bounds__((HOUT / 16) * 32)
void sage_gemm(const float* __restrict__ A1, const float* __restrict__ rowscale,
               const float* __restrict__ A2,
               const float* __restrict__ W1, const float* __restrict__ W2,
               const float* __restrict__ bias, float* __restrict__ Y) {
  __shared__ float lds[(TWO ? 2 : 1) * 16 * DIM];
  const int T   = (HOUT / 16) * 32;
  const int tid = threadIdx.x;
  const int m0  = blockIdx.x * 16;

  // stage 16xDIM tiles of A1 (scaled) and optionally A2 into LDS
  for (int i = tid; i < 16 * DIM / 4; i += T) {
    int row = i >> 5;            // 32 float4 per row of 128
    int c4  = (i & 31) * 4;
    float sc = rowscale ? rowscale[m0 + row] : 1.0f;
    v4f v = *(const v4f*)(A1 + (size_t)(m0 + row) * DIM + c4);
    v.x *= sc; v.y *= sc; v.z *= sc; v.w *= sc;
    *(v4f*)&lds[row * DIM + c4] = v;
    if (TWO) {
      v4f u = *(const v4f*)(A2 + (size_t)(m0 + row) * DIM + c4);
      *(v4f*)&lds[16 * DIM + row * DIM + c4] = u;
    }
  }
  __syncthreads();

  const int lane = tid & 31;
  const int wv   = tid >> 5;
  const int n0   = wv * 16;
  const int row  = lane & 15;    // M index for A-frag / N index for B-frag
  const int g    = lane >> 4;    // lane-group selects K pair (ISA 16x4 layout)

  v8f acc = {};
  const float* w1p = W1 + (size_t)(n0 + row) * DIM + 2 * g;
  const float* w2p = TWO ? (W2 + (size_t)(n0 + row) * DIM + 2 * g) : nullptr;
  const float* a1p = &lds[row * DIM + 2 * g];

  for (int k0 = 0; k0 < DIM; k0 += 4) {
    v2f a = *(const v2f*)(a1p + k0);          // A[row][k0+2g], A[row][k0+2g+1]
    v2f b = *(const v2f*)(w1p + k0);          // W1[n0+row][k0+2g], +1
    acc = __builtin_amdgcn_wmma_f32_16x16x4_f32(false, a, false, b,
                                                (short)0, acc, false, false);
    if (TWO) {
      v2f a2 = *(const v2f*)(a1p + 16 * DIM + k0);
      v2f b2 = *(const v2f*)(w2p + k0);
      acc = __builtin_amdgcn_wmma_f32_16x16x4_f32(false, a2, false, b2,
                                                  (short)0, acc, false, false);
    }
  }

  // C/D layout: VGPR v -> M = v + 8*g, N = lane&15
  float bb = bias[n0 + row];
  float* yp = Y + (size_t)(m0 + 8 * g) * HOUT + n0 + row;
#pragma unroll
  for (int v = 0; v < 8; ++v) {
    yp[(size_t)v * HOUT] = acc[v] + bb;
  }
}

// ------------------------------------------------------------- batch norm
__global__ void bn_stats(const float* __restrict__ Y, float* __restrict__ st) {
  int c = threadIdx.x;  // 128 channels
  float s = 0.0f, s2 = 0.0f;
  for (int r = blockIdx.x; r < NN; r += gridDim.x) {
    float v = Y[(size_t)r * DIM + c];
    s += v;
    s2 += v * v;
  }
  unsafeAtomicAdd(&st[c], s);
  unsafeAtomicAdd(&st[DIM + c], s2);
}

__global__ void bn_finalize(const float* __restrict__ st,
                            const float* __restrict__ gma,
                            const float* __restrict__ bta,
                            float* __restrict__ ss) {
  int c = threadIdx.x;
  float m   = st[c] * (1.0f / NN);
  float var = st[DIM + c] * (1.0f / NN) - m * m;
  float sc  = gma[c] * rsqrtf(var + BN_EPS);
  ss[c]       = sc;
  ss[DIM + c] = bta[c] - m * sc;
}

__global__ void bn_relu_apply(float* __restrict__ Y, const float* __restrict__ ss) {
  int i = blockIdx.x * blockDim.x + threadIdx.x;
  if (i < NN * DIM) {
    int c = i & (DIM - 1);
    float v = Y[i] * ss[c] + ss[DIM + c];
    Y[i] = v > 0.0f ? v : 0.0f;
  }
}

// ---------------------------------------------------------------- driver
extern "C" void kernel_launch(void* const* d_in, const int* in_sizes, int n_in,
                              void* d_out, int out_size, void* d_ws, size_t ws_size,
                              hipStream_t stream) {
  const float* x   = (const float*)d_in[0];
  const int*   ei  = (const int*)d_in[1];
  const float* w1l = (const float*)d_in[2];
  const float* w1r = (const float*)d_in[3];
  const float* b1c = (const float*)d_in[4];
  const float* w2l = (const float*)d_in[5];
  const float* w2r = (const float*)d_in[6];
  const float* b2c = (const float*)d_in[7];
  const float* w3l = (const float*)d_in[8];
  const float* w3r = (const float*)d_in[9];
  const float* b3c = (const float*)d_in[10];
  const float* g1  = (const float*)d_in[11];
  const float* be1 = (const float*)d_in[12];
  const float* g2  = (const float*)d_in[13];
  const float* be2 = (const float*)d_in[14];
  const float* g3  = (const float*)d_in[15];
  const float* be3 = (const float*)d_in[16];
  const float* wh  = (const float*)d_in[17];
  const float* bh  = (const float*)d_in[18];

  const int* src = ei;
  const int* dst = ei + NEDGE;

  // workspace carve (256B aligned)
  char* ws = (char*)d_ws;
  size_t off = 0;
  auto carve = [&](size_t bytes) {
    void* p = ws + off;
    off += (bytes + 255) & ~(size_t)255;
    return p;
  };
  float* agg    = (float*)carve((size_t)NN * DIM * 4);
  float* hA     = (float*)carve((size_t)NN * DIM * 4);
  float* hB     = (float*)carve((size_t)NN * DIM * 4);
  float* invd   = (float*)carve((size_t)NN * 4);
  int*   cnt    = (int*)carve((size_t)NN * 4);
  float* st     = (float*)carve(256 * 4);
  float* ss     = (float*)carve(256 * 4);

  const int ZT = 256;
  const int featN = NN * DIM;

  // degree counts (edge-only dependence, recomputed each call for determinism)
  zero_f32<<<(NN + ZT - 1) / ZT, ZT, 0, stream>>>((float*)cnt, NN);
  count_deg<<<(NEDGE + ZT - 1) / ZT, ZT, 0, stream>>>(dst, cnt, NEDGE);
  inv_deg<<<(NN + ZT - 1) / ZT, ZT, 0, stream>>>(cnt, invd, NN);

  const int scatterBlocks = (NEDGE * 32 + ZT - 1) / ZT;

  struct Layer {
    const float* in; float* out;
    const float* wl; const float* wr; const float* bc;
    const float* g;  const float* be;
  } layers[3] = {
    { x,  hA, w1l, w1r, b1c, g1, be1 },
    { hA, hB, w2l, w2r, b2c, g2, be2 },
    { hB, hA, w3l, w3r, b3c, g3, be3 },
  };

  for (int l = 0; l < 3; ++l) {
    const Layer& L = layers[l];
    zero_f32<<<(featN + ZT - 1) / ZT, ZT, 0, stream>>>(agg, featN);
    scatter_add<<<scatterBlocks, ZT, 0, stream>>>(L.in, src, dst, agg, NEDGE);
    sage_gemm<DIM, true><<<NN / 16, 256, 0, stream>>>(agg, invd, L.in,
                                                      L.wl, L.wr, L.bc, L.out);
    zero_f32<<<1, 256, 0, stream>>>(st, 256);
    bn_stats<<<256, DIM, 0, stream>>>(L.out, st);
    bn_finalize<<<1, DIM, 0, stream>>>(st, L.g, L.be, ss);
    bn_relu_apply<<<(featN + ZT - 1) / ZT, ZT, 0, stream>>>(L.out, ss);
  }

  // head: out[N,64] = h3 @ wh^T + bh   (h3 lives in hA after layer 3)
  sage_gemm<HOUT_HD, false><<<NN / 16, 128, 0, stream>>>(hA, nullptr, nullptr,
                                                         wh, nullptr, bh,
                                                         (float*)d_out);
}